// DifferentiableExtrusion_21268678050249
// MI455X (gfx1250) — compile-verified
//
#include <hip/hip_runtime.h>

typedef __attribute__((ext_vector_type(2))) float v2f;
typedef __attribute__((ext_vector_type(8))) float v8f;

#define BATCH 8
#define NPOLY 32
#define PVERT 16
#define VOX   64
#define MPTS  (VOX * VOX)            // 4096 grid points
#define EDGES (BATCH * NPOLY * PVERT) // 4096 edges total

// ---------------------------------------------------------------------------
// Kernel 1: per-edge parameters + per-polygon gates  ->  workspace (SoA)
// ws layout (floats): p1x[E] p1y[E] ux[E] uy[E] len[E] c1[E] q[E] gates[B*N]
// ---------------------------------------------------------------------------
__global__ __launch_bounds__(256) void extrude_prep(
    const float* __restrict__ polygons,       // [B,N,P,2]
    const float* __restrict__ validity,       // [B,N]
    float* __restrict__ ws) {
  int tid = blockIdx.x * 256 + threadIdx.x;   // one thread per edge
  if (tid < EDGES) {
    int pidx  = tid & (PVERT - 1);
    int nbase = tid & ~(PVERT - 1);
    int nxt   = nbase | ((pidx + 1) & (PVERT - 1));
    float p1x = polygons[2 * tid + 0] * 63.0f;
    float p1y = polygons[2 * tid + 1] * 63.0f;
    float p2x = polygons[2 * nxt + 0] * 63.0f;
    float p2y = polygons[2 * nxt + 1] * 63.0f;
    float vx = p2x - p1x, vy = p2y - p1y;
    float len = sqrtf(vx * vx + vy * vy);
    float inv = 1.0f / fmaxf(len, 1e-12f);
    float ux = vx * inv, uy = vy * inv;
    ws[0 * EDGES + tid] = p1x;
    ws[1 * EDGES + tid] = p1y;
    ws[2 * EDGES + tid] = ux;
    ws[3 * EDGES + tid] = uy;
    ws[4 * EDGES + tid] = len;
    ws[5 * EDGES + tid] = p1x * ux + p1y * uy;   // c1 = p1 . u
    ws[6 * EDGES + tid] = p1x * p1x + p1y * p1y; // q  = |p1|^2
  }
  if (tid < BATCH * NPOLY) {                     // one thread per polygon: gate
    bool nz = false;
    const float* pp = polygons + tid * PVERT * 2;
    #pragma unroll
    for (int j = 0; j < PVERT * 2; ++j) nz |= (pp[j] != 0.0f);
    bool valid = validity[tid] >= 0.5f;
    ws[7 * EDGES + tid] = (valid && nz) ? 1.0f : 0.0f;
  }
}

// ---------------------------------------------------------------------------
// Kernel 2: WMMA distance field + sigmoid + gated max + z-extrusion store.
// grid = B*32 blocks, 256 threads (8 waves). Wave w owns 16 consecutive grid
// points; the block owns 128 consecutive points; per polygon one column tile
// of its 16 edges -> two V_WMMA_F32_16X16X4_F32 (p.p1 and p.u dot products).
// ---------------------------------------------------------------------------
__global__ __launch_bounds__(256) void extrude_main(
    const float* __restrict__ ws,
    const float* __restrict__ attributes,      // [B,6]
    float* __restrict__ out) {                 // [B,64,64,64]
  const float* p1xA = ws + 0 * EDGES;
  const float* p1yA = ws + 1 * EDGES;
  const float* uxA  = ws + 2 * EDGES;
  const float* uyA  = ws + 3 * EDGES;
  const float* lenA = ws + 4 * EDGES;
  const float* c1A  = ws + 5 * EDGES;
  const float* qA   = ws + 6 * EDGES;
  const float* gate = ws + 7 * EDGES;

  __shared__ float smask[128];

  const int b    = blockIdx.x >> 5;            // 32 blocks per batch
  const int blkm = (blockIdx.x & 31) * 128;    // block's first grid point
  const int wave = threadIdx.x >> 5;
  const int lane = threadIdx.x & 31;
  const int half = lane >> 4;                  // WMMA lane half (K=0,1 vs 2,3)
  const int col  = lane & 15;                  // B/D column index (edge)
  const int m0   = blkm + wave * 16;           // wave's first grid point

  // A fragment: 16x4 fp32, rows = points. Lanes 0-15 carry K=0,1 = (x,y);
  // lanes 16-31 carry K=2,3 which we zero-pad (K is really 2).
  v2f afrag; afrag[0] = 0.0f; afrag[1] = 0.0f;
  if (half == 0) {
    int m = m0 + col;
    afrag[0] = (float)(m & 63);
    afrag[1] = (float)(m >> 6);
  }

  // |p|^2 for the 8 rows this lane sees in D (row = m0 + v + 8*half).
  float pp[8];
  #pragma unroll
  for (int v = 0; v < 8; ++v) {
    int m = m0 + v + half * 8;
    float x = (float)(m & 63), y = (float)(m >> 6);
    pp[v] = x * x + y * y;
  }

  float acc[8];
  #pragma unroll
  for (int v = 0; v < 8; ++v) acc[v] = 0.0f;

  const int ebase0 = b * (NPOLY * PVERT);
  for (int n = 0; n < NPOLY; ++n) {
    const int e = ebase0 + n * PVERT + col;    // this lane's edge (column)
    v2f bp; bp[0] = 0.0f; bp[1] = 0.0f;        // B fragment: columns = p1
    v2f bu; bu[0] = 0.0f; bu[1] = 0.0f;        // B fragment: columns = u
    if (half == 0) {
      bp[0] = p1xA[e]; bp[1] = p1yA[e];
      bu[0] = uxA[e];  bu[1] = uyA[e];
    }
    const float c1 = c1A[e];
    const float L  = lenA[e];
    const float q  = qA[e];
    const float g  = gate[b * NPOLY + n];

    v8f cz = {};
    // D1[m,e] = p_m . p1_e ; D2[m,e] = p_m . u_e  (full fp32 matrix pipe)
    v8f d1 = __builtin_amdgcn_wmma_f32_16x16x4_f32(
        false, afrag, false, bp, (short)0, cz, false, false);
    v8f d2 = __builtin_amdgcn_wmma_f32_16x16x4_f32(
        false, afrag, false, bu, (short)0, cz, false, false);

    #pragma unroll
    for (int v = 0; v < 8; ++v) {
      float s  = d2[v] - c1;                   // unclamped projection
      float t  = fminf(fmaxf(s, 0.0f), L);     // clamp to segment
      float dd = pp[v] - 2.0f * d1[v] + q + t * (t - 2.0f * s);
      float dist = sqrtf(fmaxf(dd, 0.0f));
      // min over the polygon's 16 edges = 16-lane tree within each half
      dist = fminf(dist, __shfl_xor(dist, 1, 32));
      dist = fminf(dist, __shfl_xor(dist, 2, 32));
      dist = fminf(dist, __shfl_xor(dist, 4, 32));
      dist = fminf(dist, __shfl_xor(dist, 8, 32));
      float mask = g * (1.0f / (1.0f + __expf(dist)));  // sigmoid(-d), sigma=1
      acc[v] = fmaxf(acc[v], mask);
    }
  }

  // Stage the wave's 16 mask values into LDS (values uniform per 16-lane half)
  if (col == 0) {
    #pragma unroll
    for (int v = 0; v < 8; ++v)
      smask[wave * 16 + half * 8 + v] = acc[v];
  }
  __syncthreads();

  // z-extrusion: 256 threads cover 128 points x 2 z-planes per step, coalesced
  const float h = fminf(floorf(attributes[b * 6 + 0] * (float)VOX), (float)VOX);
  const int p  = threadIdx.x & 127;
  const int z0 = threadIdx.x >> 7;
  const float val = smask[p];
  size_t obase = (size_t)b * VOX * MPTS + (size_t)(blkm + p);
  #pragma unroll 4
  for (int z = z0; z < VOX; z += 2) {
    out[obase + (size_t)z * MPTS] = ((float)z < h) ? val : 0.0f;
  }
}

extern "C" void kernel_launch(void* const* d_in, const int* in_sizes, int n_in,
                              void* d_out, int out_size, void* d_ws, size_t ws_size,
                              hipStream_t stream) {
  const float* polygons   = (const float*)d_in[0];  // [8,32,16,2]
  const float* attributes = (const float*)d_in[1];  // [8,6]
  const float* validity   = (const float*)d_in[2];  // [8,32]
  float* out = (float*)d_out;                       // [8,64,64,64]
  float* ws  = (float*)d_ws;                        // 7*4096 + 256 floats

  extrude_prep<<<EDGES / 256, 256, 0, stream>>>(polygons, validity, ws);
  extrude_main<<<BATCH * 32, 256, 0, stream>>>(ws, attributes, out);
}